// ResampleLayer_old_76716705841643
// MI455X (gfx1250) — compile-verified
//
#include <hip/hip_runtime.h>

// ResampleLayer: batched searchsorted + linear interpolation.
// B=32, S=4096, D=256, OUT_LEN=1001, duration=100, resolution=0.1.
// Memory-bound (~100 MB traffic -> ~4.3us at 23.3 TB/s roofline). Strategy:
//  - TDM (tensor_load_to_lds) stages the 16KB ts row into LDS (async, one
//    issue per block), so the 12-step binary search chases LDS, not HBM.
//  - b128 coalesced gathers/stores, nontemporal (single-pass stream).
//  - global_prefetch_b8 on next iteration's rows.

#define S_DIM 4096
#define D_DIM 256
#define B_DIM 32
#define L_OUT 1001
#define LP 64          // output timestamps per block
#define NTHREADS 256   // 8 wave32s
#define ROWS_PAR (NTHREADS / 64)  // 4 timestamps processed concurrently

typedef unsigned int u32x4 __attribute__((ext_vector_type(4)));
typedef int          i32x4 __attribute__((ext_vector_type(4)));
typedef int          i32x8 __attribute__((ext_vector_type(8)));
typedef float        f32x4 __attribute__((ext_vector_type(4)));

__global__ __launch_bounds__(NTHREADS)
void resample_interp_kernel(const float* __restrict__ embed,
                            const float* __restrict__ ts,
                            float* __restrict__ out) {
    __shared__ float sh_ts[S_DIM];   // 16 KB: one batch's timestamp row
    __shared__ int   sh_idx[LP];
    __shared__ float sh_w[LP];

    const int b   = blockIdx.y;
    const int l0  = blockIdx.x * LP;
    const int lp  = (L_OUT - l0 < LP) ? (L_OUT - l0) : LP;
    const int tid = threadIdx.x;

    const float* ts_row = ts + (size_t)b * S_DIM;

    // ---- Stage ts row into LDS via the Tensor Data Mover (CDNA5) ----
#if __has_builtin(__builtin_amdgcn_tensor_load_to_lds)
    if (tid < 32) {  // one TDM issue per block (wave 0 only; EXEC ignored)
        unsigned long long ga  = (unsigned long long)ts_row;
        unsigned int lds_byte  = (unsigned int)(unsigned long long)(&sh_ts[0]);

        // D# group 0 (CDNA5 ISA)
        u32x4 g0;
        g0.x = 1u;                                        // count=1, user mode
        g0.y = lds_byte;                                  // lds_addr [63:32]
        g0.z = (unsigned int)(ga & 0xFFFFFFFFull);        // global_addr[31:0]
        g0.w = (unsigned int)((ga >> 32) & 0x01FFFFFFull) // global_addr[56:32]
               | (2u << 30);                              // type=2 ("image")

        // D# group 1 (§8.4): data_size=4B; tensor_dim0=tile_dim0=4096;
        // tensor_dim1=tile_dim1=1; strides=4096.
        i32x8 g1;
        g1[0] = (int)(2u << 16);          // [17:16] data_size = 2 -> 4 bytes
        g1[1] = (int)(S_DIM << 16);       // tensor_dim0 low16 @ [63:48]
        g1[2] = (int)(1u << 16);          // tensor_dim0 hi=0; tensor_dim1 lo16=1
        g1[3] = (int)(S_DIM << 16);       // tensor_dim1 hi=0; tile_dim0 @ [127:112]
        g1[4] = 1;                        // tile_dim1=1, tile_dim2=0
        g1[5] = S_DIM;                    // tensor_dim0_stride low 32
        g1[6] = (int)((unsigned)S_DIM << 16); // dim0_stride hi=0; dim1_stride lo16
        g1[7] = 0;                        // tensor_dim1_stride hi

        i32x4 g2 = {0, 0, 0, 0};          // dims 2-4 unused
        i32x4 g3 = {0, 0, 0, 0};
#if __clang_major__ >= 23
        i32x8 g23 = {0, 0, 0, 0, 0, 0, 0, 0};
        __builtin_amdgcn_tensor_load_to_lds(g0, g1, g2, g3, g23, 0);
#else
        __builtin_amdgcn_tensor_load_to_lds(g0, g1, g2, g3, 0);
#endif
        __builtin_amdgcn_s_wait_tensorcnt(0);
    }
    __syncthreads();
#else
    // Fallback: cooperative b128 copy into LDS
    for (int i = tid; i < S_DIM / 4; i += NTHREADS) {
        ((f32x4*)sh_ts)[i] = ((const f32x4*)ts_row)[i];
    }
    __syncthreads();
#endif

    // ---- Binary search (searchsorted, side='left') in LDS ----
    if (tid < lp) {
        const int   l = l0 + tid;
        const float t = (float)l * 0.1f;   // linspace(0,100,1001) step
        int lo = 0, hi = S_DIM;
        while (lo < hi) {
            int mid = (lo + hi) >> 1;
            if (sh_ts[mid] < t) lo = mid + 1; else hi = mid;
        }
        int idx = lo;
        idx = (idx < 1) ? 1 : idx;
        idx = (idx > S_DIM - 1) ? (S_DIM - 1) : idx;
        const float x_lo = sh_ts[idx - 1];
        const float x_hi = sh_ts[idx];
        sh_idx[tid] = idx;
        sh_w[tid]   = (t - x_lo) / (x_hi - x_lo);
    }
    __syncthreads();

    // ---- Gather + interpolate: 64 lanes x f32x4 per timestamp ----
    const int    li0    = tid >> 6;          // 0..3
    const int    ch     = (tid & 63) * 4;    // channel base (x4 floats)
    const size_t base_b = (size_t)b * S_DIM * D_DIM;

    for (int li = li0; li < lp; li += ROWS_PAR) {
        const int   idx = sh_idx[li];
        const float w   = sh_w[li];

        // Prefetch next iteration's rows (global_prefetch_b8)
        if (li + ROWS_PAR < lp) {
            const int nidx = sh_idx[li + ROWS_PAR];
            __builtin_prefetch(embed + base_b + (size_t)(nidx - 1) * D_DIM + ch, 0, 0);
            __builtin_prefetch(embed + base_b + (size_t)nidx       * D_DIM + ch, 0, 0);
        }

        // Rows are read ~once (P(reuse) ~ e^-4.1): nontemporal b128 gathers.
        const f32x4 ylo = __builtin_nontemporal_load(
            (const f32x4*)(embed + base_b + (size_t)(idx - 1) * D_DIM + ch));
        const f32x4 yhi = __builtin_nontemporal_load(
            (const f32x4*)(embed + base_b + (size_t)idx       * D_DIM + ch));

        f32x4 r;
        r.x = __builtin_fmaf(yhi.x - ylo.x, w, ylo.x);
        r.y = __builtin_fmaf(yhi.y - ylo.y, w, ylo.y);
        r.z = __builtin_fmaf(yhi.z - ylo.z, w, ylo.z);
        r.w = __builtin_fmaf(yhi.w - ylo.w, w, ylo.w);

        // Write-once output stream: nontemporal b128 store.
        __builtin_nontemporal_store(
            r, (f32x4*)(out + ((size_t)b * L_OUT + (l0 + li)) * D_DIM + ch));
    }
}

extern "C" void kernel_launch(void* const* d_in, const int* in_sizes, int n_in,
                              void* d_out, int out_size, void* d_ws, size_t ws_size,
                              hipStream_t stream) {
    const float* embed = (const float*)d_in[0];  // [B, S, D] float32
    const float* ts    = (const float*)d_in[1];  // [B, S]    float32 (sorted)
    float*       out   = (float*)d_out;          // [B, L_OUT, D] float32

    dim3 grid((L_OUT + LP - 1) / LP, B_DIM);     // (16, 32) blocks
    resample_interp_kernel<<<grid, NTHREADS, 0, stream>>>(embed, ts, out);
}